// GcnInfluence_52003464020718
// MI455X (gfx1250) — compile-verified
//
#include <hip/hip_runtime.h>
#include <hip/hip_bf16.h>
#include <math.h>

#define N_NODES 100000
#define N_FEAT  512
#define N_EDGES 3200000
#define SCAN_B  256
#define N_SCANB ((N_NODES + SCAN_B - 1) / SCAN_B)   // 391

typedef __attribute__((ext_vector_type(2))) float v2f;
typedef __attribute__((ext_vector_type(8))) float v8f;

// ---------------------------------------------------------------- utilities
__global__ void k_zero_i32(int* __restrict__ p, int n) {
    int i = blockIdx.x * blockDim.x + threadIdx.x;
    if (i < n) p[i] = 0;
}

// cnt[dst] += 1 over all edges
__global__ void k_count(const int* __restrict__ ei, int* __restrict__ cnt) {
    int e = blockIdx.x * blockDim.x + threadIdx.x;
    if (e < N_EDGES) atomicAdd(&cnt[ei[N_EDGES + e]], 1);
}

// dinv[i] = rsqrt(in_degree + self_loop)
__global__ void k_dinv(const int* __restrict__ cnt, float* __restrict__ dinv) {
    int i = blockIdx.x * blockDim.x + threadIdx.x;
    if (i < N_NODES) dinv[i] = rsqrtf((float)cnt[i] + 1.0f);
}

// ---------------------------------------------------------------- scan (CSR build)
// Per-block inclusive scan of cnt -> row (temp), block totals -> bsum
__global__ void k_scan_block(const int* __restrict__ cnt, int* __restrict__ row,
                             int* __restrict__ bsum, int n) {
    __shared__ int sh[SCAN_B];
    const int i = blockIdx.x * SCAN_B + threadIdx.x;
    int v = (i < n) ? cnt[i] : 0;
    sh[threadIdx.x] = v;
    __syncthreads();
    for (int off = 1; off < SCAN_B; off <<= 1) {
        int t = (threadIdx.x >= off) ? sh[threadIdx.x - off] : 0;
        __syncthreads();
        sh[threadIdx.x] += t;
        __syncthreads();
    }
    if (i < n) row[i] = sh[threadIdx.x];
    if (threadIdx.x == SCAN_B - 1) bsum[blockIdx.x] = sh[SCAN_B - 1];
}

// Serial exclusive scan of block sums (391 elements; negligible)
__global__ void k_scan_sums(int* __restrict__ bsum, int nb) {
    if (blockIdx.x == 0 && threadIdx.x == 0) {
        int acc = 0;
        for (int b = 0; b < nb; ++b) { int t = bsum[b]; bsum[b] = acc; acc += t; }
    }
}

// row[i] := exclusive scan; cursor := copy; row[N] := E
__global__ void k_scan_finish(const int* __restrict__ cnt, int* __restrict__ row,
                              const int* __restrict__ bsum, int* __restrict__ cursor,
                              int n) {
    int i = blockIdx.x * blockDim.x + threadIdx.x;
    if (i < n) {
        int ex = row[i] - cnt[i] + bsum[i / SCAN_B];
        row[i] = ex;
        cursor[i] = ex;
    }
    if (i == 0) row[n] = N_EDGES;
}

// srcs[pos] = src, bucketed by dst
__global__ void k_bucket(const int* __restrict__ ei, int* __restrict__ cursor,
                         int* __restrict__ srcs) {
    int e = blockIdx.x * blockDim.x + threadIdx.x;
    if (e < N_EDGES) {
        int s = ei[e];
        int d = ei[N_EDGES + e];
        int pos = atomicAdd(&cursor[d], 1);
        srcs[pos] = s;
    }
}

// ---------------------------------------------------------------- GEMM (WMMA f32)
// One wave per 16x64 output stripe, K stepped by 4 via V_WMMA_F32_16X16X4_F32.
__global__ void __launch_bounds__(32)
k_gemm_xw(const float* __restrict__ X, const float* __restrict__ W,
          float* __restrict__ XW) {
    const int lane = threadIdx.x & 31;
    const int half = lane >> 4;        // 0 | 1
    const int lm   = lane & 15;        // 0..15
    const int tm   = blockIdx.x;       // 16-row tile   (6250 tiles)
    const int tn   = blockIdx.y;       // 64-col stripe (8 stripes)

    const float* __restrict__ xrow = X + (size_t)(tm * 16 + lm) * N_FEAT;
    const int c0 = tn * 64 + lm;

    v8f acc0 = {}, acc1 = {}, acc2 = {}, acc3 = {};

    for (int k = 0; k < N_FEAT; k += 4) {
        v2f a;
        a.x = xrow[k + 2 * half + 0];
        a.y = xrow[k + 2 * half + 1];

        const float* __restrict__ w0 = W + (size_t)(k + half) * N_FEAT;
        const float* __restrict__ w2 = W + (size_t)(k + 2 + half) * N_FEAT;

        v2f b0, b1, b2, b3;
        b0.x = w0[c0 +  0]; b0.y = w2[c0 +  0];
        b1.x = w0[c0 + 16]; b1.y = w2[c0 + 16];
        b2.x = w0[c0 + 32]; b2.y = w2[c0 + 32];
        b3.x = w0[c0 + 48]; b3.y = w2[c0 + 48];

        acc0 = __builtin_amdgcn_wmma_f32_16x16x4_f32(false, a, false, b0, (short)0, acc0, false, false);
        acc1 = __builtin_amdgcn_wmma_f32_16x16x4_f32(false, a, false, b1, (short)0, acc1, false, false);
        acc2 = __builtin_amdgcn_wmma_f32_16x16x4_f32(false, a, false, b2, (short)0, acc2, false, false);
        acc3 = __builtin_amdgcn_wmma_f32_16x16x4_f32(false, a, false, b3, (short)0, acc3, false, false);
    }

    #pragma unroll
    for (int v = 0; v < 8; ++v) {
        const size_t r = (size_t)(tm * 16 + v + 8 * half) * N_FEAT + c0;
        XW[r +  0] = acc0[v];
        XW[r + 16] = acc1[v];
        XW[r + 32] = acc2[v];
        XW[r + 48] = acc3[v];
    }
}

// ---------------------------------------------------------------- CSR aggregate + finalize
// One wave32 per node: acc = sum_{e in row} dinv[src]*dinv[node]*XW[src]
//                        + dinv[node]^2 * XW[node]           (self loop)
// then bias + relu + softmax, single store of the output row.
// Lane feature map: feature(j,t) = j*128 + lane*4 + t  (float4-coalesced)
__global__ void __launch_bounds__(256)
k_aggregate(const int* __restrict__ row, const int* __restrict__ srcs,
            const float* __restrict__ xw, const float* __restrict__ dinv,
            const float* __restrict__ bias, float* __restrict__ out) {
    const int node = blockIdx.x * (blockDim.x >> 5) + (threadIdx.x >> 5);
    const int lane = threadIdx.x & 31;
    if (node >= N_NODES) return;

    const float di = dinv[node];
    float acc[16];

    // self-loop seed
    {
        const float4* __restrict__ xr = (const float4*)(xw + (size_t)node * N_FEAT);
        const float w = di * di;
        #pragma unroll
        for (int j = 0; j < 4; ++j) {
            float4 v = xr[j * 32 + lane];
            acc[j*4+0] = v.x * w; acc[j*4+1] = v.y * w;
            acc[j*4+2] = v.z * w; acc[j*4+3] = v.w * w;
        }
    }

    const int e0 = row[node], e1 = row[node + 1];
    for (int e = e0; e < e1; ++e) {
        const int s = srcs[e];
        if (e + 1 < e1) {  // hide random-row latency (global_prefetch_b8)
            const int sn = srcs[e + 1];
            __builtin_prefetch((const char*)(xw + (size_t)sn * N_FEAT) + lane * 16, 0, 1);
        }
        const float w = dinv[s] * di;
        const float4* __restrict__ xr = (const float4*)(xw + (size_t)s * N_FEAT);
        #pragma unroll
        for (int j = 0; j < 4; ++j) {
            float4 v = xr[j * 32 + lane];
            acc[j*4+0] += v.x * w; acc[j*4+1] += v.y * w;
            acc[j*4+2] += v.z * w; acc[j*4+3] += v.w * w;
        }
    }

    // bias + relu, track max
    float mx = -3.402823466e38f;
    #pragma unroll
    for (int j = 0; j < 4; ++j)
        #pragma unroll
        for (int t = 0; t < 4; ++t) {
            const int f = j * 128 + lane * 4 + t;
            float x = fmaxf(acc[j*4+t] + bias[f], 0.0f);
            acc[j*4+t] = x;
            mx = fmaxf(mx, x);
        }
    #pragma unroll
    for (int off = 16; off >= 1; off >>= 1)
        mx = fmaxf(mx, __shfl_xor(mx, off, 32));

    float s = 0.0f;
    #pragma unroll
    for (int q = 0; q < 16; ++q) {
        acc[q] = __expf(acc[q] - mx);
        s += acc[q];
    }
    #pragma unroll
    for (int off = 16; off >= 1; off >>= 1)
        s += __shfl_xor(s, off, 32);

    const float inv = 1.0f / s;
    float4* __restrict__ orow = (float4*)(out + (size_t)node * N_FEAT);
    #pragma unroll
    for (int j = 0; j < 4; ++j) {
        float4 v;
        v.x = acc[j*4+0] * inv; v.y = acc[j*4+1] * inv;
        v.z = acc[j*4+2] * inv; v.w = acc[j*4+3] * inv;
        orow[j * 32 + lane] = v;
    }
}

// ---------------------------------------------------------------- launcher
extern "C" void kernel_launch(void* const* d_in, const int* in_sizes, int n_in,
                              void* d_out, int out_size, void* d_ws, size_t ws_size,
                              hipStream_t stream) {
    const float* X  = (const float*)d_in[0];   // [100000, 512]
    const int*   EI = (const int*)d_in[1];     // [2, 3200000]
    const float* W  = (const float*)d_in[2];   // [512, 512]
    const float* B  = (const float*)d_in[3];   // [512]
    float* out = (float*)d_out;                // [100000, 512]

    // workspace carve
    float* xw     = (float*)d_ws;                    // 51,200,000 f
    float* dinv   = xw + (size_t)N_NODES * N_FEAT;   // 100,000 f
    int*   cnt    = (int*)(dinv + N_NODES);          // 100,000
    int*   row    = cnt + N_NODES;                   // 100,001
    int*   cursor = row + N_NODES + 1;               // 100,000
    int*   bsum   = cursor + N_NODES;                // 512 (N_SCANB used)
    int*   srcs   = bsum + 512;                      // 3,200,000

    // 1) dense GEMM XW = X @ W (f32 WMMA)
    {
        dim3 grid(N_NODES / 16, N_FEAT / 64);        // 6250 x 8
        k_gemm_xw<<<grid, 32, 0, stream>>>(X, W, xw);
    }

    // 2) CSR build: count, dinv, scan, bucket
    k_zero_i32<<<(N_NODES + 255) / 256, 256, 0, stream>>>(cnt, N_NODES);
    k_count<<<(N_EDGES + 255) / 256, 256, 0, stream>>>(EI, cnt);
    k_dinv<<<(N_NODES + 255) / 256, 256, 0, stream>>>(cnt, dinv);
    k_scan_block<<<N_SCANB, SCAN_B, 0, stream>>>(cnt, row, bsum, N_NODES);
    k_scan_sums<<<1, 32, 0, stream>>>(bsum, N_SCANB);
    k_scan_finish<<<(N_NODES + 255) / 256, 256, 0, stream>>>(cnt, row, bsum, cursor, N_NODES);
    k_bucket<<<(N_EDGES + 255) / 256, 256, 0, stream>>>(EI, cursor, srcs);

    // 3) atomic-free aggregation fused with bias+relu+softmax
    k_aggregate<<<(N_NODES + 7) / 8, 256, 0, stream>>>(row, srcs, xw, dinv, B, out);
}